// HistLoss_72464688218854
// MI455X (gfx1250) — compile-verified
//
#include <hip/hip_runtime.h>
#include <hip/hip_bf16.h>

// ---------------------------------------------------------------------------
// HistLoss for MI455X (gfx1250, wave32).
// S=2 styles, C=256 channels, N=65536 elems/channel, 256 bins.
//
// Kernel 1: CDFs for all 512 (s,c) rows via fp32 WMMA triangular matmul:
//           cdf_raw[512x256] = t_hist[512x256] @ L (L[i][j] = i<=j).
// Kernel 2: per-(s,c) block: masked load -> LDS (256KB), min/max, LDS float
//           histogram, in-LDS bitonic sort, rank->bin via binary search over
//           scaled CDF, MSE accumulation into d_out[0].
// ---------------------------------------------------------------------------

#define SNUM 2
#define CNUM 256
#define NELEM 65536      // H*W
#define NBINS 256
#define NROWS (SNUM * CNUM)   // 512
#define TPB 1024

typedef float v2f __attribute__((ext_vector_type(2)));
typedef float v8f __attribute__((ext_vector_type(8)));

// ----------------------------------------------------------------------------
// Kernel 1: cdf_raw[r][j] = sum_{i<=j} t_hist[r][i]  via V_WMMA_F32_16X16X4_F32
// One wave = one 16x16 output tile. 512 tiles total (32 row-tiles x 16 col).
// Also zeroes the loss accumulator d_out[0].
// ----------------------------------------------------------------------------
__global__ void __launch_bounds__(256) cdf_wmma_kernel(
    const float* __restrict__ thist,   // [512][256]
    float* __restrict__ cdf_raw,       // [512][256] (workspace)
    float* __restrict__ out)           // d_out (out[0] = loss accumulator)
{
  if (blockIdx.x == 0 && threadIdx.x == 0) out[0] = 0.0f;

  const int wave = (blockIdx.x * blockDim.x + threadIdx.x) >> 5;  // 0..511
  const int lane = threadIdx.x & 31;
  const int rowTile = wave >> 4;   // 0..31  -> rows R0..R0+15
  const int colTile = wave & 15;   // 0..15  -> cols B0..B0+15
  const int R0 = rowTile << 4;
  const int B0 = colTile << 4;

  // A (16x4 fp32): lanes 0-15 -> M=lane, K={0,1}; lanes 16-31 -> M=lane-16, K={2,3}
  const int m  = lane & 15;
  const int kb = (lane >> 4) << 1;           // 0 or 2
  const float* __restrict__ arow = thist + (size_t)(R0 + m) * NBINS;

  v8f acc = {};
  for (int k = 0; k < NBINS; k += 4) {
    v2f a;
    a.x = arow[k + kb];
    a.y = arow[k + kb + 1];
    // B (4x16): L[i][j] = (i <= j), generated branchlessly.
    v2f b;
    b.x = ((k + kb)     <= (B0 + m)) ? 1.0f : 0.0f;
    b.y = ((k + kb + 1) <= (B0 + m)) ? 1.0f : 0.0f;
    acc = __builtin_amdgcn_wmma_f32_16x16x4_f32(
        /*neg_a=*/false, a, /*neg_b=*/false, b,
        /*c_mod=*/(short)0, acc, /*reuse_a=*/false, /*reuse_b=*/false);
  }

  // D (16x16 fp32): VGPR p -> M = p + 8*(lane>=16), N = lane&15.
  const int nCol  = B0 + (lane & 15);
  const int rBase = R0 + ((lane >> 4) << 3);
#pragma unroll
  for (int p = 0; p < 8; ++p)
    cdf_raw[(size_t)(rBase + p) * NBINS + nCol] = acc[p];
}

// ----------------------------------------------------------------------------
// Kernel 2: one block per (s,c). Whole channel lives in LDS.
// ----------------------------------------------------------------------------
__global__ void __launch_bounds__(TPB, 1) hist_sort_loss_kernel(
    const float* __restrict__ input,   // [C][N] (batch dim 1 dropped)
    const float* __restrict__ masks,   // [S][N]
    const float* __restrict__ cdf_raw, // [512][256]
    const float* __restrict__ tmins,   // [512]
    const float* __restrict__ tmaxs,   // [512]
    float* __restrict__ out)           // [1 + 512*256]
{
  extern __shared__ __align__(16) float vals[];   // 65536 floats = 256 KB
  __shared__ float cdf_s[NBINS];
  __shared__ float hist_s[NBINS];
  __shared__ float rmin[TPB];
  __shared__ float rmax[TPB];
  __shared__ float blk_acc;

  const int r   = blockIdx.x;       // 0..511
  const int s   = r >> 8;
  const int c   = r & 255;
  const int tid = threadIdx.x;

  const float4* __restrict__ xch = (const float4*)(input + (size_t)c * NELEM);
  const float4* __restrict__ msk = (const float4*)(masks + (size_t)s * NELEM);

  // ---- masked load -> LDS, per-thread min/max -------------------------------
  float lmin =  3.4e38f;
  float lmax = -3.4e38f;
  for (int i = tid; i < NELEM / 4; i += TPB) {
    if (i + TPB < NELEM / 4) {
      __builtin_prefetch(&xch[i + TPB], 0, 0);   // global_prefetch_b8
      __builtin_prefetch(&msk[i + TPB], 0, 0);
    }
    float4 xv = xch[i];
    float4 mv = msk[i];
    float4 v = make_float4(xv.x * mv.x, xv.y * mv.y, xv.z * mv.z, xv.w * mv.w);
    ((float4*)vals)[i] = v;
    lmin = fminf(lmin, fminf(fminf(v.x, v.y), fminf(v.z, v.w)));
    lmax = fmaxf(lmax, fmaxf(fmaxf(v.x, v.y), fmaxf(v.z, v.w)));
  }
  rmin[tid] = lmin;
  rmax[tid] = lmax;
  if (tid < NBINS) hist_s[tid] = 0.0f;
  if (tid == 0) blk_acc = 0.0f;
  __syncthreads();

  // ---- block min/max tree reduction ----------------------------------------
  for (int off = TPB / 2; off > 0; off >>= 1) {
    if (tid < off) {
      rmin[tid] = fminf(rmin[tid], rmin[tid + off]);
      rmax[tid] = fmaxf(rmax[tid], rmax[tid + off]);
    }
    __syncthreads();
  }
  const float lo = rmin[0];
  const float hi = rmax[0];
  const float scale = (float)(NBINS - 1) / fmaxf(hi - lo, 1e-12f);

  // ---- histogram (ds_add_f32 atomics) --------------------------------------
  for (int i = tid; i < NELEM; i += TPB) {
    float v = vals[i];
    int b = (int)floorf((v - lo) * scale + 0.5f);
    b = min(max(b, 0), NBINS - 1);
    atomicAdd(&hist_s[b], 1.0f);
  }

  // ---- load + scale cdf: cdf = raw_cumsum * (N / max(total, 1e-12)) --------
  if (tid < NBINS) {
    float total = cdf_raw[(size_t)r * NBINS + (NBINS - 1)];
    float csc = (float)NELEM / fmaxf(total, 1e-12f);
    cdf_s[tid] = cdf_raw[(size_t)r * NBINS + tid] * csc;
  }
  __syncthreads();
  if (tid < NBINS) out[1 + (size_t)r * NBINS + tid] = hist_s[tid];

  // ---- in-LDS bitonic sort of the 65536 values -----------------------------
  for (unsigned k = 2; k <= (unsigned)NELEM; k <<= 1) {
    for (unsigned j = k >> 1; j > 0; j >>= 1) {
      for (unsigned i = tid; i < (unsigned)NELEM; i += TPB) {
        unsigned ixj = i ^ j;
        if (ixj > i) {
          float a = vals[i];
          float b = vals[ixj];
          bool up = ((i & k) == 0u);
          if ((a > b) == up) { vals[i] = b; vals[ixj] = a; }
        }
      }
      __syncthreads();
    }
  }

  // ---- loss: sum_i (sorted[i] - f(rank i))^2 -------------------------------
  // f(i) = clip(lower_bound(cdf, i+0.5), 0, 255)/255 * (tmax - tmin) + tmin
  const float tmn = tmins[r];
  const float rng = tmaxs[r] - tmn;
  float acc = 0.0f;
  for (int i = tid; i < NELEM; i += TPB) {
    const float target = (float)i + 0.5f;
    int l = 0, h = NBINS;                 // lower_bound over cdf_s
    while (l < h) {
      int mid = (l + h) >> 1;
      if (cdf_s[mid] < target) l = mid + 1; else h = mid;
    }
    int bin = min(l, NBINS - 1);
    float t = (float)bin * (1.0f / (float)(NBINS - 1)) * rng + tmn;
    float d = vals[i] - t;
    acc += d * d;
  }
  atomicAdd(&blk_acc, acc);
  __syncthreads();
  if (tid == 0) {
    // loss = sum_s mean_{C,N}; 0.01*strength == 1.0
    atomicAdd(out, blk_acc * (1.0f / (float)(CNUM * NELEM)));
  }
}

// ----------------------------------------------------------------------------
extern "C" void kernel_launch(void* const* d_in, const int* in_sizes, int n_in,
                              void* d_out, int out_size, void* d_ws, size_t ws_size,
                              hipStream_t stream) {
  (void)in_sizes; (void)n_in; (void)out_size; (void)ws_size;
  const float* input = (const float*)d_in[0];   // [1,256,256,256]
  const float* masks = (const float*)d_in[1];   // [2,256,256]
  const float* thist = (const float*)d_in[2];   // [2,256,256]
  const float* tmins = (const float*)d_in[3];   // [2,256]
  const float* tmaxs = (const float*)d_in[4];   // [2,256]
  float* out = (float*)d_out;                   // [1 + 2*256*256]
  float* cdf_raw = (float*)d_ws;                // 512*256 floats = 512 KB

  // 512 waves, one 16x16 WMMA tile each (also zeroes out[0]).
  cdf_wmma_kernel<<<64, 256, 0, stream>>>(thist, cdf_raw, out);

  // 256 KB dynamic LDS per workgroup (whole channel resident in LDS).
  hipFuncSetAttribute((const void*)hist_sort_loss_kernel,
                      hipFuncAttributeMaxDynamicSharedMemorySize,
                      NELEM * (int)sizeof(float));
  hist_sort_loss_kernel<<<NROWS, TPB, NELEM * sizeof(float), stream>>>(
      input, masks, cdf_raw, tmins, tmaxs, out);
}